// BehaviorPlant_30296699306436
// MI455X (gfx1250) — compile-verified
//
#include <hip/hip_runtime.h>
#include <stdint.h>

// Problem constants (from reference)
#define BATCH 32
#define CHN   20
#define HH    256
#define WW    256

#define EMPTY_C 0
#define WATER_C 3
#define WOOD_C  5
#define ICE_C   6
#define PLANT_C 8

typedef float v4f __attribute__((ext_vector_type(4)));
typedef __attribute__((address_space(1))) int gint_t;  // global int
typedef __attribute__((address_space(3))) int lint_t;  // LDS int

// Async global -> LDS copy (CDNA5 ASYNCcnt path). Builtin signature on this
// toolchain: (global int*, local int*, imm offset, imm cpol).
__device__ __forceinline__ void async_cp_f32(const float* g, float* l) {
#if __has_builtin(__builtin_amdgcn_global_load_async_to_lds_b32)
  __builtin_amdgcn_global_load_async_to_lds_b32((gint_t*)g, (lint_t*)l,
                                                /*imm offset*/0, /*cpol*/0);
#else
  *l = *g;
#endif
}

__device__ __forceinline__ void wait_async_lds() {
#if __has_builtin(__builtin_amdgcn_s_wait_asynccnt)
  __builtin_amdgcn_s_wait_asynccnt(0);
#else
  asm volatile("s_wait_asynccnt 0" ::: "memory");
#endif
}

// One block = 4 rows x 256 cols of one batch image. 256 threads = 8 wave32.
// Thread t: ty = t>>6 (row in tile), tx = t&63 (float4 strip of 4 columns).
__global__ __launch_bounds__(256)
void behavior_plant_kernel(const float* __restrict__ world,
                           const float* __restrict__ rand_interact,
                           const float* __restrict__ ev_plant,
                           const float* __restrict__ ev_empty,
                           float* __restrict__ out) {
  __shared__ float sP[6][WW];  // PLANT rows h0-1 .. h0+4
  __shared__ float sW[6][WW];  // WOOD
  __shared__ float sI[6][WW];  // ICE

  const int t  = threadIdx.x;
  const int b  = blockIdx.y;
  const int h0 = blockIdx.x * 4;
  const size_t cstride = (size_t)HH * WW;   // per-channel stride

  // ---- Stage stencil halo rows into LDS via async DMA (zero-pad OOB rows) ----
#pragma unroll
  for (int r = 0; r < 6; ++r) {
    const int hg = h0 - 1 + r;
    if (hg >= 0 && hg < HH) {
      const size_t base = (((size_t)b * CHN) * HH + (size_t)hg) * WW + t;
      async_cp_f32(world + base + (size_t)PLANT_C * cstride, &sP[r][t]);
      async_cp_f32(world + base + (size_t)WOOD_C  * cstride, &sW[r][t]);
      async_cp_f32(world + base + (size_t)ICE_C   * cstride, &sI[r][t]);
    } else {
      sP[r][t] = 0.0f;
      sW[r][t] = 0.0f;
      sI[r][t] = 0.0f;
    }
  }
  wait_async_lds();
  __syncthreads();

  // ---- Per-thread tile coordinates ----
  const int tx = t & 63;
  const int ty = t >> 6;
  const int cb = tx * 4;        // first of 4 columns owned by this thread
  const int h  = h0 + ty;

  // ---- 3x3 box sums from LDS: vertical column sums first, then horizontal ----
  float colP[6], colWI[6];
#pragma unroll
  for (int j = 0; j < 6; ++j) {
    const int c = cb - 1 + j;
    if (c >= 0 && c < WW) {
      colP[j]  = sP[ty][c] + sP[ty + 1][c] + sP[ty + 2][c];
      colWI[j] = (sW[ty][c] + sI[ty][c]) +
                 (sW[ty + 1][c] + sI[ty + 1][c]) +
                 (sW[ty + 2][c] + sI[ty + 2][c]);
    } else {
      colP[j]  = 0.0f;
      colWI[j] = 0.0f;
    }
  }

  // ---- Per-pixel predicate inputs (coalesced row loads) ----
  const float* wpix = world + ((size_t)b * CHN) * cstride + (size_t)h * WW + cb;
  const size_t pix  = ((size_t)b * HH + h) * WW + cb;

  const v4f wtr = *(const v4f*)(wpix + (size_t)WATER_C * cstride);
  const v4f emp = *(const v4f*)(wpix + (size_t)EMPTY_C * cstride);
  const v4f ri  = *(const v4f*)(rand_interact + pix);

  v4f FA, FB, FN;
#pragma unroll
  for (int p = 0; p < 4; ++p) {
    const float pcv = colP[p] + colP[p + 1] + colP[p + 2];    // plant_counts
    const float wic = colWI[p] + colWI[p + 1] + colWI[p + 2]; // wood_ice_counts
    const bool water = wtr[p] > 0.5f;
    const bool empty = emp[p] > 0.5f;
    const float rc   = ri[p];
    const bool grow  = water && (rc < 0.05f);
    bool aP = grow && (pcv <= 3.0f) && (pcv >= 1.0f);
    const bool aE = grow && (pcv > 3.0f);
    aP = aP || ((wic > 0.0f) && (rc < 0.2f) && empty && (pcv > 0.0f));
    FA[p] = aP ? 1.0f : 0.0f;
    FB[p] = aE ? 1.0f : 0.0f;
    FN[p] = (!aP && !aE) ? 1.0f : 0.0f;
  }

  // ---- Stream all 20 channels: NT b128 load, blend, NT b128 store ----
  float* opix = out + ((size_t)b * CHN) * cstride + (size_t)h * WW + cb;
#pragma unroll 4
  for (int c = 0; c < CHN; ++c) {
    const float pv = ev_plant[c];   // uniform -> scalar loads
    const float ev = ev_empty[c];
    const v4f wv = __builtin_nontemporal_load((const v4f*)(wpix + (size_t)c * cstride));
    const v4f o  = FN * wv + FA * pv + FB * ev;
    __builtin_nontemporal_store(o, (v4f*)(opix + (size_t)c * cstride));
  }
}

extern "C" void kernel_launch(void* const* d_in, const int* in_sizes, int n_in,
                              void* d_out, int out_size, void* d_ws, size_t ws_size,
                              hipStream_t stream) {
  (void)in_sizes; (void)n_in; (void)out_size; (void)d_ws; (void)ws_size;
  // setup_inputs() order:
  // 0: world [32,20,256,256] f32
  // 1: rand_movement (unused by reference)
  // 2: rand_interact [32,1,256,256] f32
  // 3: rand_element (unused)
  // 4: neighbor_kernel (all-ones 3x3, baked into stencil)
  // 5: elem_vec_plant [1,20,1,1] f32
  // 6: elem_vec_empty [1,20,1,1] f32
  const float* world = (const float*)d_in[0];
  const float* rand_interact = (const float*)d_in[2];
  const float* ev_plant = (const float*)d_in[5];
  const float* ev_empty = (const float*)d_in[6];
  float* out = (float*)d_out;

  dim3 grid(HH / 4, BATCH);   // 64 x 32 blocks
  dim3 block(256);            // 8 wave32 per block, 18 KB LDS
  behavior_plant_kernel<<<grid, block, 0, stream>>>(world, rand_interact,
                                                    ev_plant, ev_empty, out);
}